// Net_82429012345018
// MI455X (gfx1250) — compile-verified
//
#include <hip/hip_runtime.h>
#include <math.h>

// ---------------------------------------------------------------------------
// GCN forward on gfx1250 (CDNA5): f32 WMMA GEMMs (LDS-staged pair-interleaved
// weights, one A-fragment reused across all N tiles) + L2-resident atomic
// edge aggregation.
// ---------------------------------------------------------------------------

typedef __attribute__((ext_vector_type(2))) float v2f;
typedef __attribute__((ext_vector_type(8))) float v8f;

// ---------------- degree / normalization kernels ----------------

__global__ void deg_init_kernel(float* __restrict__ deg, int n) {
    int i = blockIdx.x * blockDim.x + threadIdx.x;
    if (i < n) deg[i] = 1.0f;  // self-loop contribution
}

__global__ void deg_accum_kernel(const int* __restrict__ dst, float* __restrict__ deg, int e) {
    int i = blockIdx.x * blockDim.x + threadIdx.x;
    if (i < e) atomicAdd(&deg[dst[i]], 1.0f);
}

__global__ void rsqrt_inplace_kernel(float* __restrict__ d, int n) {
    int i = blockIdx.x * blockDim.x + threadIdx.x;
    if (i < n) d[i] = rsqrtf(d[i]);  // deg >= 1 always (self-loops)
}

__global__ void edge_norm_kernel(const int* __restrict__ src, const int* __restrict__ dst,
                                 const float* __restrict__ dinv, float* __restrict__ nrm, int e) {
    int i = blockIdx.x * blockDim.x + threadIdx.x;
    if (i < e) nrm[i] = dinv[src[i]] * dinv[dst[i]];
}

// ---------------- WMMA GEMM: Hlin[N x NCOLS] = H[N x KDIM] @ W[KDIM x NCOLS] --------------
// Weights staged to LDS zero-padded to TN*16 cols and K-PAIR interleaved:
//   sW[kp * NP + c] = { W[2kp][c], W[2kp+1][c] }  (aligned float2)
// so every B fragment is one ds_load_b64 straight into an adjacent VGPR pair.
// One wave owns a 16-row strip: per 4-wide k-step it loads ONE A fragment
// (global_load_b64) and reuses it across TN WMMAs. Requires n_rows % 16 == 0
// (true: 100000 = 6250*16). EXEC is all-ones around every WMMA.

template <int KDIM, int NCOLS>
__global__ __launch_bounds__(256) void gemm_wmma_f32(const float* __restrict__ H,
                                                     const float* __restrict__ Wm,
                                                     float* __restrict__ Hlin,
                                                     int n_rows) {
    constexpr int TN = (NCOLS + 15) / 16;   // N tiles: 4 (64) or 3 (40)
    constexpr int NP = TN * 16;             // padded cols: 64 or 48
    constexpr int KP = KDIM / 2;            // k-pairs
    __shared__ v2f sW[KP * NP];

    // cooperative, zero-padded, pair-interleaved stage of W into LDS
    for (int idx = threadIdx.x; idx < KP * NP; idx += 256) {
        int kp  = idx / NP;
        int col = idx - kp * NP;
        v2f val = {0.0f, 0.0f};
        if (col < NCOLS) {
            val.x = Wm[(2 * kp) * NCOLS + col];
            val.y = Wm[(2 * kp + 1) * NCOLS + col];
        }
        sW[idx] = val;
    }
    __syncthreads();

    const int wave = blockIdx.x * 8 + (threadIdx.x >> 5);
    const int lane = threadIdx.x & 31;
    const int tiles_m = n_rows >> 4;
    if (wave >= tiles_m) return;  // wave-uniform

    // A fragment: row = tm*16 + (lane&15); lane<16 -> K = k,k+1 ; lane>=16 -> K = k+2,k+3
    const int mrow = wave * 16 + (lane & 15);
    const int koff = (lane >> 4) << 1;  // 0 or 2
    const float* __restrict__ arow = H + (size_t)mrow * KDIM + koff;

    // B fragment: pair-row kp = k/2 + (lane>=16), column = tn*16 + (lane&15)
    const v2f* __restrict__ brow = sW + (size_t)(koff >> 1) * NP + (lane & 15);

    v8f acc[TN];
#pragma unroll
    for (int t = 0; t < TN; ++t) acc[t] = v8f{};

#pragma unroll
    for (int k = 0; k < KDIM; k += 4) {
        v2f a = *(const v2f*)(arow + k);                 // global_load_b64
        const v2f* bk = brow + (size_t)(k >> 1) * NP;    // immediate-offset ds_load_b64s
#pragma unroll
        for (int t = 0; t < TN; ++t) {
            v2f b = bk[t * 16];
            acc[t] = __builtin_amdgcn_wmma_f32_16x16x4_f32(false, a, false, b, (short)0,
                                                           acc[t], false, false);
        }
    }

    // C/D layout: VGPR r -> (M = r or r+8 for upper half-lanes), N = lane&15
    const int base_m = wave * 16 + ((lane >> 4) << 3);
#pragma unroll
    for (int t = 0; t < TN; ++t) {
        int c = t * 16 + (lane & 15);
        if (c < NCOLS) {  // compile-time true for NCOLS=64; masks only last tile for 40
#pragma unroll
            for (int r = 0; r < 8; ++r)
                Hlin[(size_t)(base_m + r) * NCOLS + c] = acc[t][r];
        }
    }
}

// ---------------- aggregation: self-loop + bias init, then atomic edge scatter ----------------

template <int FEAT>
__global__ void init_agg_kernel(const float* __restrict__ lin, const float* __restrict__ dinv,
                                const float* __restrict__ bias, float* __restrict__ agg,
                                int n) {
    int i = blockIdx.x * blockDim.x + threadIdx.x;
    if (i >= n * FEAT) return;
    int node = i / FEAT;  // constant divisor -> strength-reduced
    int f = i - node * FEAT;
    float di = dinv[node];
    agg[i] = lin[i] * (di * di) + bias[f];
}

// blockDim = (64, 4): 64 feature lanes per edge (coalesced gather + L2 atomics)
template <int FEAT>
__global__ void edge_agg_kernel(const int* __restrict__ src, const int* __restrict__ dst,
                                const float* __restrict__ nrm, const float* __restrict__ lin,
                                float* __restrict__ agg, int n_edges) {
    int f = threadIdx.x;
    int e = blockIdx.x * blockDim.y + threadIdx.y;
    if (e >= n_edges || f >= FEAT) return;  // f-guard compile-time dead for FEAT=64
    int s = src[e];
    int d = dst[e];
    float v = lin[(size_t)s * FEAT + f] * nrm[e];
    atomicAdd(&agg[(size_t)d * FEAT + f], v);
}

__global__ void silu_kernel(float* __restrict__ h, int n) {
    int i = blockIdx.x * blockDim.x + threadIdx.x;
    if (i < n) {
        float x = h[i];
        h[i] = x / (1.0f + expf(-x));
    }
}

// ---------------- log-softmax: one wave32 per row (feat <= 64) ----------------

__global__ void log_softmax_kernel(const float* __restrict__ in, float* __restrict__ out,
                                   int n_rows, int feat) {
    int row = blockIdx.x * blockDim.y + threadIdx.y;
    if (row >= n_rows) return;
    int lane = threadIdx.x;  // 0..31, wave32
    const float* rp = in + (size_t)row * feat;

    float v0 = (lane < feat) ? rp[lane] : -INFINITY;
    float v1 = (lane + 32 < feat) ? rp[lane + 32] : -INFINITY;

    float m = fmaxf(v0, v1);
#pragma unroll
    for (int off = 16; off > 0; off >>= 1) m = fmaxf(m, __shfl_xor(m, off, 32));

    float s = ((lane < feat) ? expf(v0 - m) : 0.0f) +
              ((lane + 32 < feat) ? expf(v1 - m) : 0.0f);
#pragma unroll
    for (int off = 16; off > 0; off >>= 1) s += __shfl_xor(s, off, 32);

    float lse = m + logf(s);
    float* op = out + (size_t)row * feat;
    if (lane < feat) op[lane] = v0 - lse;
    if (lane + 32 < feat) op[lane + 32] = v1 - lse;
}

// ---------------------------------------------------------------------------

extern "C" void kernel_launch(void* const* d_in, const int* in_sizes, int n_in,
                              void* d_out, int out_size, void* d_ws, size_t ws_size,
                              hipStream_t stream) {
    const float* x  = (const float*)d_in[0];
    const int*   ei = (const int*)d_in[1];
    const float* W1 = (const float*)d_in[2];
    const float* b1 = (const float*)d_in[3];
    const float* W2 = (const float*)d_in[4];
    const float* b2 = (const float*)d_in[5];
    const float* W3 = (const float*)d_in[6];
    const float* b3 = (const float*)d_in[7];
    const float* W4 = (const float*)d_in[8];
    const float* b4 = (const float*)d_in[9];
    float* out = (float*)d_out;

    const int N = in_sizes[0] / 128;  // 100000 (divisible by 16)
    const int E = in_sizes[1] / 2;    // 1600000
    const int* src = ei;
    const int* dst = ei + E;

    // workspace layout (floats): dinv[N] | norm[E] | bufL[N*64] | bufA[N*64] | bufB[N*64]
    float* ws   = (float*)d_ws;
    float* dinv = ws;
    float* nrm  = dinv + N;
    float* bufL = nrm + E;
    float* bufA = bufL + (size_t)N * 64;
    float* bufB = bufA + (size_t)N * 64;

    // --- normalization: deg = 1 + indeg ; dinv = rsqrt(deg) ; norm_e = dinv[s]*dinv[d]
    deg_init_kernel<<<(N + 255) / 256, 256, 0, stream>>>(dinv, N);
    deg_accum_kernel<<<(E + 255) / 256, 256, 0, stream>>>(dst, dinv, E);
    rsqrt_inplace_kernel<<<(N + 255) / 256, 256, 0, stream>>>(dinv, N);
    edge_norm_kernel<<<(E + 255) / 256, 256, 0, stream>>>(src, dst, dinv, nrm, E);

    const int tiles_m = N / 16;
    const int gemm_blocks = (tiles_m + 7) / 8;  // 8 waves per 256-thread block
    const dim3 eb(64, 4);

    // layer 1: x(128) -> 64
    gemm_wmma_f32<128, 64><<<gemm_blocks, 256, 0, stream>>>(x, W1, bufL, N);
    init_agg_kernel<64><<<(N * 64 + 255) / 256, 256, 0, stream>>>(bufL, dinv, b1, bufA, N);
    edge_agg_kernel<64><<<(E + 3) / 4, eb, 0, stream>>>(src, dst, nrm, bufL, bufA, E);
    silu_kernel<<<(N * 64 + 255) / 256, 256, 0, stream>>>(bufA, N * 64);

    // layer 2: 64 -> 64
    gemm_wmma_f32<64, 64><<<gemm_blocks, 256, 0, stream>>>(bufA, W2, bufL, N);
    init_agg_kernel<64><<<(N * 64 + 255) / 256, 256, 0, stream>>>(bufL, dinv, b2, bufB, N);
    edge_agg_kernel<64><<<(E + 3) / 4, eb, 0, stream>>>(src, dst, nrm, bufL, bufB, E);
    silu_kernel<<<(N * 64 + 255) / 256, 256, 0, stream>>>(bufB, N * 64);

    // layer 3: 64 -> 64
    gemm_wmma_f32<64, 64><<<gemm_blocks, 256, 0, stream>>>(bufB, W3, bufL, N);
    init_agg_kernel<64><<<(N * 64 + 255) / 256, 256, 0, stream>>>(bufL, dinv, b3, bufA, N);
    edge_agg_kernel<64><<<(E + 3) / 4, eb, 0, stream>>>(src, dst, nrm, bufL, bufA, E);
    silu_kernel<<<(N * 64 + 255) / 256, 256, 0, stream>>>(bufA, N * 64);

    // layer 4: 64 -> 40
    gemm_wmma_f32<64, 40><<<gemm_blocks, 256, 0, stream>>>(bufA, W4, bufL, N);
    init_agg_kernel<40><<<(N * 40 + 255) / 256, 256, 0, stream>>>(bufL, dinv, b4, bufB, N);
    edge_agg_kernel<40><<<(E + 3) / 4, eb, 0, stream>>>(src, dst, nrm, bufL, bufB, E);

    dim3 lb(32, 8);
    log_softmax_kernel<<<(N + 7) / 8, lb, 0, stream>>>(bufB, out, N, 40);
}